// MoE_10204842295759
// MI455X (gfx1250) — compile-verified
//
#include <hip/hip_runtime.h>
#include <math.h>

typedef __attribute__((ext_vector_type(2))) float v2f;
typedef __attribute__((ext_vector_type(8))) float v8f;

#define DIM 1024
#define HID 2816
#define NEXP 8
#define BM 32            // tokens per block tile
#define BH 128           // H-chunk
#define NTHREADS 256     // 8 wave32

#define WMMA_F32X4(A, B, C) \
  __builtin_amdgcn_wmma_f32_16x16x4_f32(false, (A), false, (B), (short)0, (C), false, false)

__device__ inline v8f vzero8() {
  v8f v;
  #pragma unroll
  for (int i = 0; i < 8; ++i) v[i] = 0.0f;
  return v;
}

// ---------------- Router: logits -> top2 -> softmax -> dense cw[N,8] ----------------
__global__ __launch_bounds__(256) void moe_router(const float* __restrict__ x,
                                                  const float* __restrict__ Wgate,
                                                  float* __restrict__ cw, int N) {
  int n = blockIdx.x * blockDim.x + threadIdx.x;
  if (n >= N) return;
  const float* xr = x + (size_t)n * DIM;
  float acc[NEXP];
  #pragma unroll
  for (int e = 0; e < NEXP; ++e) acc[e] = 0.0f;
  for (int d = 0; d < DIM; ++d) {
    float xv = xr[d];
    const float* wr = Wgate + d * NEXP;
    #pragma unroll
    for (int e = 0; e < NEXP; ++e) acc[e] += xv * wr[e];
  }
  // top-2, ties -> lower index (matches jax.lax.top_k)
  int i0 = 0; float m0 = acc[0];
  #pragma unroll
  for (int e = 1; e < NEXP; ++e) { if (acc[e] > m0) { m0 = acc[e]; i0 = e; } }
  int i1 = (i0 == 0) ? 1 : 0; float m1 = acc[i1];
  #pragma unroll
  for (int e = 0; e < NEXP; ++e) {
    if (e == i0) continue;
    if (acc[e] > m1) { m1 = acc[e]; i1 = e; }
  }
  float e1 = __expf(m1 - m0);
  float w0 = 1.0f / (1.0f + e1);
  float w1 = e1 * w0;
  float* o = cw + (size_t)n * NEXP;
  #pragma unroll
  for (int e = 0; e < NEXP; ++e) o[e] = 0.0f;
  o[i0] = w0;
  o[i1] = w1;
}

// ---------------- Fused SwiGLU MoE FFN with fp32 WMMA ----------------
__global__ __launch_bounds__(NTHREADS) void moe_ffn(const float* __restrict__ x,
                                                    const float* __restrict__ Wg,
                                                    const float* __restrict__ Wu,
                                                    const float* __restrict__ Wd,
                                                    const float* __restrict__ cw,
                                                    float* __restrict__ out) {
  __shared__ float sx[BM * DIM];    // 128 KB: x tile, resident for whole block
  __shared__ float sh[BM * BH];     //  16 KB: current h chunk (cw-scaled)
  __shared__ float scw[BM * NEXP];  //   1 KB
  __shared__ int   sflags[NEXP];

  const int tid   = threadIdx.x;
  const int wave  = tid >> 5;       // 0..7 (wave32)
  const int lane  = tid & 31;
  const int lhalf = lane >> 4;      // 0/1
  const int l16   = lane & 15;
  const int row0  = blockIdx.x * BM;

  // load x tile + cw tile (coalesced); compute per-expert "used in this tile" flags once
  for (int i = tid; i < BM * DIM; i += NTHREADS) sx[i] = x[(size_t)row0 * DIM + i];
  if (tid < BM * NEXP) scw[tid] = cw[(size_t)row0 * NEXP + tid];
  if (tid < NEXP) sflags[tid] = 0;
  __syncthreads();
  if (tid < BM * NEXP) { if (scw[tid] != 0.0f) atomicOr(&sflags[tid & (NEXP - 1)], 1); }
  __syncthreads();

  // out accumulators: wave owns cols [wave*128, wave*128+128) => 2 Mtiles x 8 Ntiles
  v8f acc[2][8];
  #pragma unroll
  for (int mt = 0; mt < 2; ++mt)
    #pragma unroll
    for (int nt = 0; nt < 8; ++nt) acc[mt][nt] = vzero8();

  for (int e = 0; e < NEXP; ++e) {
    if (sflags[e] == 0) continue;   // block-uniform, deterministic skip

    // per-lane routing weights for the rows this lane owns in C tiles
    float cwr[2][8];
    #pragma unroll
    for (int mt = 0; mt < 2; ++mt)
      #pragma unroll
      for (int r = 0; r < 8; ++r)
        cwr[mt][r] = scw[(mt * 16 + 8 * lhalf + r) * NEXP + e];

    const float* wgE = Wg + (size_t)e * DIM * HID;
    const float* wuE = Wu + (size_t)e * DIM * HID;
    const float* wdE = Wd + (size_t)e * HID * DIM;

    for (int hc = 0; hc < HID / BH; ++hc) {
      // ---- Phase A: h[:, chunk] ; wave computes cols [wave*16, +16) ----
      const int hcol = hc * BH + wave * 16 + l16;
      v8f cg0 = vzero8(), cg1 = vzero8(), cu0 = vzero8(), cu1 = vzero8();
      #pragma unroll 4
      for (int kb = 0; kb < DIM; kb += 4) {
        const int k0 = kb + 2 * lhalf;                 // A/B frag: k = k0 + j
        v2f a0 = *(const v2f*)&sx[l16 * DIM + k0];         // rows 0..15
        v2f a1 = *(const v2f*)&sx[(16 + l16) * DIM + k0];  // rows 16..31
        v2f bg, bu;
        bg.x = wgE[(size_t)k0 * HID + hcol];
        bg.y = wgE[(size_t)(k0 + 1) * HID + hcol];
        bu.x = wuE[(size_t)k0 * HID + hcol];
        bu.y = wuE[(size_t)(k0 + 1) * HID + hcol];
        cg0 = WMMA_F32X4(a0, bg, cg0);
        cg1 = WMMA_F32X4(a1, bg, cg1);
        cu0 = WMMA_F32X4(a0, bu, cu0);
        cu1 = WMMA_F32X4(a1, bu, cu1);
      }

      __syncthreads();  // prior chunk's h fully consumed before overwrite
      #pragma unroll
      for (int r = 0; r < 8; ++r) {
        float g0 = cg0[r], u0 = cu0[r];
        float s0 = __builtin_amdgcn_rcpf(1.0f + __expf(-g0));   // sigmoid via v_rcp_f32
        sh[(8 * lhalf + r) * BH + wave * 16 + l16] = g0 * s0 * u0 * cwr[0][r];
        float g1 = cg1[r], u1 = cu1[r];
        float s1 = __builtin_amdgcn_rcpf(1.0f + __expf(-g1));
        sh[(16 + 8 * lhalf + r) * BH + wave * 16 + l16] = g1 * s1 * u1 * cwr[1][r];
      }
      __syncthreads();

      // ---- Phase B: acc += h_chunk @ Wd[chunk, wave's 128 cols] ----
      const int ncol0 = wave * 128;
      #pragma unroll 2
      for (int kb = 0; kb < BH; kb += 4) {
        const int k0 = kb + 2 * lhalf;
        v2f a0 = *(const v2f*)&sh[l16 * BH + k0];
        v2f a1 = *(const v2f*)&sh[(16 + l16) * BH + k0];
        const float* wd0 = wdE + (size_t)(hc * BH + k0) * DIM;
        const float* wd1 = wd0 + DIM;
        #pragma unroll
        for (int nt = 0; nt < 8; ++nt) {
          v2f b;
          b.x = wd0[ncol0 + nt * 16 + l16];
          b.y = wd1[ncol0 + nt * 16 + l16];
          acc[0][nt] = WMMA_F32X4(a0, b, acc[0][nt]);
          acc[1][nt] = WMMA_F32X4(a1, b, acc[1][nt]);
        }
      }
    }
  }

  // ---- store out tile: row = r + 8*lhalf (+16*mt), col = wave*128 + nt*16 + l16 ----
  #pragma unroll
  for (int mt = 0; mt < 2; ++mt)
    #pragma unroll
    for (int nt = 0; nt < 8; ++nt)
      #pragma unroll
      for (int r = 0; r < 8; ++r) {
        int row = row0 + mt * 16 + 8 * lhalf + r;
        int col = wave * 128 + nt * 16 + l16;
        out[(size_t)row * DIM + col] = acc[mt][nt][r];
      }
}

extern "C" void kernel_launch(void* const* d_in, const int* in_sizes, int n_in,
                              void* d_out, int out_size, void* d_ws, size_t ws_size,
                              hipStream_t stream) {
  const float* x     = (const float*)d_in[0];
  const float* Wgate = (const float*)d_in[1];
  const float* Wg    = (const float*)d_in[2];
  const float* Wu    = (const float*)d_in[3];
  const float* Wd    = (const float*)d_in[4];
  float* out = (float*)d_out;
  const int N = in_sizes[0] / DIM;     // 8192 tokens

  float* cw = (float*)d_ws;            // N*8 floats, fully overwritten each call

  moe_router<<<(N + 255) / 256, 256, 0, stream>>>(x, Wgate, cw, N);
  moe_ffn<<<N / BM, NTHREADS, 0, stream>>>(x, Wg, Wu, Wd, cw, out);
}